// RigidBodySimulator_63668595196396
// MI455X (gfx1250) — compile-verified
//
#include <hip/hip_runtime.h>
#include <stdint.h>

typedef __attribute__((ext_vector_type(2))) float v2f;
typedef __attribute__((ext_vector_type(8))) float v8f;

#define RB_NB     64            // persistent blocks (must all be resident)
#define RB_BLOCK  256           // 8 waves (wave32)
#define RB_NWAVES (RB_BLOCK/32)
#define RB_CHMAX  1600          // padded per-block vertex capacity (mult of 16)
#define RB_DT     (1.0f/600.0f) // 1/60/10

// workspace layout in floats
#define WS_PI     8                       // I_ref partials: NB*6
#define WS_PS     (WS_PI + RB_NB*6)       // step partials, double buffered: 2*NB*4
#define WS_FLOATS (WS_PS + 2*RB_NB*4)

#if __has_builtin(__builtin_amdgcn_wmma_f32_16x16x4_f32)
#define RB_HAVE_WMMA 1
#else
#define RB_HAVE_WMMA 0
#endif

__device__ __forceinline__ float wave_sum(float v) {
  v += __shfl_xor(v, 16, 32);
  v += __shfl_xor(v, 8, 32);
  v += __shfl_xor(v, 4, 32);
  v += __shfl_xor(v, 2, 32);
  v += __shfl_xor(v, 1, 32);
  return v;
}

// monotonic grid barrier: phase p completes when counter reaches p*NB
__device__ __forceinline__ void grid_barrier(unsigned* cnt, unsigned phase) {
  __syncthreads();
  if (threadIdx.x == 0) {
    __threadfence();
    atomicAdd(cnt, 1u);
    const unsigned target = phase * (unsigned)RB_NB;
    while (atomicAdd(cnt, 0u) < target) {
      __builtin_amdgcn_s_sleep(1);
    }
    __threadfence();
  }
  __syncthreads();
}

__device__ __forceinline__ void m3_mul(const float a[9], const float b[9], float c[9]) {
#pragma unroll
  for (int i = 0; i < 3; ++i)
#pragma unroll
    for (int j = 0; j < 3; ++j)
      c[3*i+j] = a[3*i+0]*b[0+j] + a[3*i+1]*b[3+j] + a[3*i+2]*b[6+j];
}

__device__ __forceinline__ void m3_mul_bt(const float a[9], const float b[9], float c[9]) {
#pragma unroll
  for (int i = 0; i < 3; ++i)
#pragma unroll
    for (int j = 0; j < 3; ++j)
      c[3*i+j] = a[3*i+0]*b[3*j+0] + a[3*i+1]*b[3*j+1] + a[3*i+2]*b[3*j+2];
}

__device__ __forceinline__ void m3_inv(const float a[9], float o[9]) {
  float c00 = a[4]*a[8] - a[5]*a[7];
  float c01 = a[5]*a[6] - a[3]*a[8];
  float c02 = a[3]*a[7] - a[4]*a[6];
  float det = a[0]*c00 + a[1]*c01 + a[2]*c02;
  float id  = 1.0f / det;
  o[0] = c00*id; o[1] = (a[2]*a[7]-a[1]*a[8])*id; o[2] = (a[1]*a[5]-a[2]*a[4])*id;
  o[3] = c01*id; o[4] = (a[0]*a[8]-a[2]*a[6])*id; o[5] = (a[2]*a[3]-a[0]*a[5])*id;
  o[6] = c02*id; o[7] = (a[1]*a[6]-a[0]*a[7])*id; o[8] = (a[0]*a[4]-a[1]*a[3])*id;
}

__device__ __forceinline__ void m3_vec(const float a[9], const float v[3], float o[3]) {
  o[0] = a[0]*v[0] + a[1]*v[1] + a[2]*v[2];
  o[1] = a[3]*v[0] + a[4]*v[1] + a[5]*v[2];
  o[2] = a[6]*v[0] + a[7]*v[1] + a[8]*v[2];
}

__device__ __forceinline__ void quat_to_R(float q0, float q1, float q2, float q3, float R[9]) {
  float inv = 1.0f / sqrtf(q0*q0 + q1*q1 + q2*q2 + q3*q3);
  float w = q0*inv, x = q1*inv, y = q2*inv, z = q3*inv;
  R[0] = 1.0f - 2.0f*y*y - 2.0f*z*z; R[1] = 2.0f*x*y - 2.0f*w*z;        R[2] = 2.0f*x*z + 2.0f*w*y;
  R[3] = 2.0f*x*y + 2.0f*w*z;        R[4] = 1.0f - 2.0f*x*x - 2.0f*z*z; R[5] = 2.0f*y*z - 2.0f*w*x;
  R[6] = 2.0f*x*z - 2.0f*w*y;        R[7] = 2.0f*y*z + 2.0f*w*x;        R[8] = 1.0f - 2.0f*x*x - 2.0f*y*y;
}

__global__ __launch_bounds__(RB_BLOCK)
void rb_persistent_kernel(const float* __restrict__ xg,
                          const float* __restrict__ mc_p,
                          const float* __restrict__ t0_p,
                          const float* __restrict__ q0_p,
                          const float* __restrict__ v0_p,
                          const float* __restrict__ w0_p,
                          const float* __restrict__ kn_p,
                          const float* __restrict__ mu_p,
                          const float* __restrict__ ld_p,
                          const float* __restrict__ ad_p,
                          const int*   __restrict__ steps_p,
                          float* __restrict__ out,
                          float* __restrict__ ws,
                          int N) {
  __shared__ float xs[RB_CHMAX], ys[RB_CHMAX], zs[RB_CHMAX];
  __shared__ float red[RB_NWAVES * 6];
  __shared__ float coeff[8];

  unsigned* cnt = (unsigned*)ws;
  float* pI = ws + WS_PI;
  float* pS = ws + WS_PS;

  const int tid  = threadIdx.x;
  const int blk  = blockIdx.x;
  const int lane = tid & 31;
  const int wid  = tid >> 5;
  const int col  = lane & 15;
  const bool hi  = lane >= 16;

  const int CH    = (N + RB_NB - 1) / RB_NB;   // 1563 for N=100000
  const int vbase = blk * CH;
  int cN = N - vbase;
  if (cN > CH) cN = CH;
  if (cN < 0)  cN = 0;

  const int steps = steps_p[0];
  const float mcx = mc_p[0], mcy = mc_p[1], mcz = mc_p[2];

  // ---------------- stage chunk into LDS via CDNA5 async global->LDS ----------------
  {
    unsigned long long gbase = (unsigned long long)xg;
    unsigned lx = (unsigned)(size_t)(&xs[0]);
    unsigned ly = (unsigned)(size_t)(&ys[0]);
    unsigned lz = (unsigned)(size_t)(&zs[0]);
    for (int i = tid; i < cN; i += RB_BLOCK) {
      unsigned goff = (unsigned)((vbase + i) * 12);
      unsigned l0 = lx + 4u * (unsigned)i;
      unsigned l1 = ly + 4u * (unsigned)i;
      unsigned l2 = lz + 4u * (unsigned)i;
      asm volatile("global_load_async_to_lds_b32 %0, %1, %2"
                   :: "v"(l0), "v"(goff), "s"(gbase) : "memory");
      asm volatile("global_load_async_to_lds_b32 %0, %1, %2 offset:4"
                   :: "v"(l1), "v"(goff), "s"(gbase) : "memory");
      asm volatile("global_load_async_to_lds_b32 %0, %1, %2 offset:8"
                   :: "v"(l2), "v"(goff), "s"(gbase) : "memory");
    }
    for (int i = cN + tid; i < RB_CHMAX; i += RB_BLOCK) { xs[i] = 0.f; ys[i] = 0.f; zs[i] = 0.f; }
    asm volatile("s_wait_asynccnt 0x0" ::: "memory");
    __syncthreads();
  }

  // ---------------- phase 1: I_ref partial sums (Σ outer(r,r), r = x - mc) ----------------
  {
    float s0=0.f,s1=0.f,s2=0.f,s3=0.f,s4=0.f,s5=0.f;
    for (int i = tid; i < cN; i += RB_BLOCK) {
      float rx = xs[i]-mcx, ry = ys[i]-mcy, rz = zs[i]-mcz;
      s0 += rx*rx; s1 += rx*ry; s2 += rx*rz; s3 += ry*ry; s4 += ry*rz; s5 += rz*rz;
    }
    s0 = wave_sum(s0); s1 = wave_sum(s1); s2 = wave_sum(s2);
    s3 = wave_sum(s3); s4 = wave_sum(s4); s5 = wave_sum(s5);
    if (lane == 0) {
      red[wid*6+0]=s0; red[wid*6+1]=s1; red[wid*6+2]=s2;
      red[wid*6+3]=s3; red[wid*6+4]=s4; red[wid*6+5]=s5;
    }
    __syncthreads();
    if (tid == 0) {
      float a[6] = {0.f,0.f,0.f,0.f,0.f,0.f};
      for (int wv = 0; wv < RB_NWAVES; ++wv)
        for (int k = 0; k < 6; ++k) a[k] += red[wv*6+k];
      for (int k = 0; k < 6; ++k) pI[blk*6+k] = a[k];
    }
  }
  grid_barrier(cnt, 1);

  // ---------------- thread 0: build I_ref, load initial state ----------------
  float Iref[9] = {0,0,0,0,0,0,0,0,0};
  float tx=0,ty=0,tz=0, q0=1,q1=0,q2=0,q3=0, vx=0,vy=0,vz=0, wx=0,wy=0,wz=0;
  float kn=0, mu=0, ld=1, ad=1, invm=0;
  if (tid == 0) {
    float S[6] = {0.f,0.f,0.f,0.f,0.f,0.f};
    for (int b = 0; b < RB_NB; ++b)
      for (int k = 0; k < 6; ++k) S[k] += pI[b*6+k];
    float tr = S[0] + S[3] + S[5];
    Iref[0]=tr-S[0]; Iref[1]=-S[1];   Iref[2]=-S[2];
    Iref[3]=-S[1];   Iref[4]=tr-S[3]; Iref[5]=-S[4];
    Iref[6]=-S[2];   Iref[7]=-S[4];   Iref[8]=tr-S[5];
    tx = t0_p[0]; ty = t0_p[1]; tz = t0_p[2];
    q0 = q0_p[0]; q1 = q0_p[1]; q2 = q0_p[2]; q3 = q0_p[3];
    float qn = 1.0f / sqrtf(q0*q0+q1*q1+q2*q2+q3*q3);
    q0*=qn; q1*=qn; q2*=qn; q3*=qn;
    vx = v0_p[0]; vy = v0_p[1]; vz = v0_p[2];
    wx = w0_p[0]; wy = w0_p[1]; wz = w0_p[2];
    kn = kn_p[0]; mu = mu_p[0]; ld = ld_p[0]; ad = ad_p[0];
    invm = 1.0f / (float)N;
  }

  const int ntiles = (cN + 15) >> 4;

  // ---------------- main 120-step loop ----------------
  for (int s = 0; s < steps; ++s) {
    float R[9];
    if (tid == 0) {
      // v = (v + DT*g)*ld ; w = w*ad
      vx = vx * ld; vy = (vy - 9.8f * RB_DT) * ld; vz = vz * ld;
      wx *= ad; wy *= ad; wz *= ad;
      quat_to_R(q0, q1, q2, q3, R);
      coeff[0] = R[3]; coeff[1] = R[4]; coeff[2] = R[5];          // d:     n·(R x)
      coeff[3] = ty + mcy;                                        // d bias
      coeff[4] = wz*R[0] - wx*R[6];                               // rel_v coeffs
      coeff[5] = wz*R[1] - wx*R[7];
      coeff[6] = wz*R[2] - wx*R[8];
      coeff[7] = vy;                                              // rel_v bias
    }
    __syncthreads();

    const float c0=coeff[0], c1=coeff[1], c2=coeff[2], c3=coeff[3];
    const float c4=coeff[4], c5=coeff[5], c6=coeff[6], c7=coeff[7];

#if RB_HAVE_WMMA
    // A (16x4 f32): row0 = (a0,a1,a2,b), row1 = (c0,c1,c2,e), rest zero.
    // lane m<16 holds A[m][0],A[m][1]; lane 16+m holds A[m][2],A[m][3].
    float A0 = 0.f, A1 = 0.f;
    if (lane == 0)       { A0 = c0; A1 = c1; }
    else if (lane == 1)  { A0 = c4; A1 = c5; }
    else if (lane == 16) { A0 = c2; A1 = c3; }
    else if (lane == 17) { A0 = c6; A1 = c7; }
    v2f Am; Am.x = A0; Am.y = A1;
    v8f Cz = {};
#endif

    float ax=0.f, ay=0.f, az=0.f, an=0.f;
    for (int tt = wid; tt < ntiles; tt += RB_NWAVES) {
      int vi = tt*16 + col;
      float xx = xs[vi], yy = ys[vi], zz = zs[vi];
      float dval, rv;
#if RB_HAVE_WMMA
      // B (4x16 f32): column n = (x,y,z,1) of vertex n.
      v2f Bm; Bm.x = hi ? zz : xx; Bm.y = hi ? 1.0f : yy;
      v8f D = __builtin_amdgcn_wmma_f32_16x16x4_f32(false, Am, false, Bm,
                                                    (short)0, Cz, false, false);
      dval = D[0];   // row 0 -> d  (lanes 0-15); high lanes: zero rows
      rv   = D[1];   // row 1 -> rel_v
#else
      dval = fmaf(c0, xx, fmaf(c1, yy, fmaf(c2, zz, c3)));
      rv   = fmaf(c4, xx, fmaf(c5, yy, fmaf(c6, zz, c7)));
#endif
      if (!hi && vi < cN && dval < 0.f && rv < 0.f) {
        ax += xx; ay += yy; az += zz; an += 1.f;
      }
    }

    ax = wave_sum(ax); ay = wave_sum(ay); az = wave_sum(az); an = wave_sum(an);
    if (lane == 0) { red[wid*6+0]=ax; red[wid*6+1]=ay; red[wid*6+2]=az; red[wid*6+3]=an; }
    __syncthreads();
    if (tid == 0) {
      float bx=0.f, by=0.f, bz=0.f, bn=0.f;
      for (int wv = 0; wv < RB_NWAVES; ++wv) {
        bx += red[wv*6+0]; by += red[wv*6+1]; bz += red[wv*6+2]; bn += red[wv*6+3];
      }
      float* dst = pS + (size_t)((s & 1) * RB_NB + blk) * 4;
      dst[0]=bx; dst[1]=by; dst[2]=bz; dst[3]=bn;
    }

    grid_barrier(cnt, (unsigned)(2 + s));

    if (tid == 0) {
      float sx=0.f, sy=0.f, sz=0.f, sn=0.f;
      const float* src = pS + (size_t)((s & 1) * RB_NB) * 4;
      for (int b = 0; b < RB_NB; ++b) {
        sx += src[b*4+0]; sy += src[b*4+1]; sz += src[b*4+2]; sn += src[b*4+3];
      }
      float fn  = fmaxf(sn, 1.0f);
      float crx = sx/fn, cry = sy/fn, crz = sz/fn;
      float Rix = R[0]*crx + R[1]*cry + R[2]*crz;
      float Riy = R[3]*crx + R[4]*cry + R[5]*crz;
      float Riz = R[6]*crx + R[7]*cry + R[8]*crz;
      float vix = vx + wy*Riz - wz*Riy;
      float viy = vy + wz*Rix - wx*Riz;
      float viz = vz + wx*Riy - wy*Rix;
      float nvn = fabsf(viy);
      float nvt = sqrtf(vix*vix + viz*viz);
      float alpha = fmaxf(0.f, 1.f - mu*(1.f+kn)*(nvn/(nvt + 1e-6f)));
      // vi_new = -kn*v_n + alpha*v_t
      float rhs[3] = { alpha*vix - vix, -kn*viy - viy, alpha*viz - viz };
      float T9[9], Im[9], Iinv[9];
      m3_mul(R, Iref, T9);
      m3_mul_bt(T9, R, Im);           // I = R Iref R^T
      m3_inv(Im, Iinv);
      float Cm[9] = { 0.f,-Riz, Riy,  Riz, 0.f,-Rix,  -Riy, Rix, 0.f };
      float U9[9], CU[9], Km[9], Kinv[9];
      m3_mul(Cm, Iinv, U9);
      m3_mul(U9, Cm, CU);             // C Iinv C
      for (int k = 0; k < 9; ++k) Km[k] = -CU[k];
      Km[0] += invm; Km[4] += invm; Km[8] += invm;
      m3_inv(Km, Kinv);
      float J[3]; m3_vec(Kinv, rhs, J);
      if (sn > 0.f) {
        vx += J[0]*invm; vy += J[1]*invm; vz += J[2]*invm;
        float CJ[3]; m3_vec(Cm, J, CJ);
        float dW[3]; m3_vec(Iinv, CJ, dW);
        wx += dW[0]; wy += dW[1]; wz += dW[2];
      }
      tx += RB_DT*vx; ty += RB_DT*vy; tz += RB_DT*vz;
      // q += 0.5*DT*quat_mul([0,w], q); normalize
      float dqw = -(wx*q1 + wy*q2 + wz*q3);
      float dqx =   wx*q0 + wy*q3 - wz*q2;
      float dqy =   wy*q0 + wz*q1 - wx*q3;
      float dqz =   wz*q0 + wx*q2 - wy*q1;
      q0 += 0.5f*RB_DT*dqw; q1 += 0.5f*RB_DT*dqx;
      q2 += 0.5f*RB_DT*dqy; q3 += 0.5f*RB_DT*dqz;
      float qn = 1.0f / sqrtf(q0*q0 + q1*q1 + q2*q2 + q3*q3);
      q0*=qn; q1*=qn; q2*=qn; q3*=qn;
      if (blk == 0) { out[3*s+0] = tx; out[3*s+1] = ty; out[3*s+2] = tz; }
    }
  }
}

extern "C" void kernel_launch(void* const* d_in, const int* in_sizes, int n_in,
                              void* d_out, int out_size, void* d_ws, size_t ws_size,
                              hipStream_t stream) {
  const float* xg = (const float*)d_in[0];
  const float* mc = (const float*)d_in[1];
  const float* t0 = (const float*)d_in[2];
  const float* q0 = (const float*)d_in[3];
  const float* v0 = (const float*)d_in[4];
  const float* w0 = (const float*)d_in[5];
  const float* kn = (const float*)d_in[6];
  const float* mu = (const float*)d_in[7];
  const float* ld = (const float*)d_in[8];
  const float* ad = (const float*)d_in[9];
  const int* steps = (const int*)d_in[10];
  const int N = in_sizes[0] / 3;

  hipMemsetAsync(d_ws, 0, WS_FLOATS * sizeof(float), stream);
  hipLaunchKernelGGL(rb_persistent_kernel, dim3(RB_NB), dim3(RB_BLOCK), 0, stream,
                     xg, mc, t0, q0, v0, w0, kn, mu, ld, ad, steps,
                     (float*)d_out, (float*)d_ws, N);
}